// FinalMLP_40827959115924
// MI455X (gfx1250) — compile-verified
//
#include <hip/hip_runtime.h>
#include <math.h>

// ---------------------------------------------------------------------------
// Equivariant MLP forward for MI455X (gfx1250, wave32).
// Linears run on fp32 matrix cores via V_WMMA_F32_16X16X4_F32.
// ---------------------------------------------------------------------------

typedef __attribute__((ext_vector_type(2))) float v2f;
typedef __attribute__((ext_vector_type(8))) float v8f;

#define NBATCH 131072
#define TANH_C 1.5927116f
#define SIG_C  1.8467354f

// ---------------------------------------------------------------------------
// GEMM: Y[n, offY + v*IR + i] = scale * sum_u X[n, offX + u*IR + i] * W[u, v]
// Rows of the GEMM are flattened (n, i) pairs: R = N*IR (always mult of 128).
// Each 256-thread block = 8 waves; wave w owns 16 rows x (NT*16) columns.
//
// Weights staged in LDS with consecutive K-row pairs interleaved:
//   element (u, c) -> lw[((u>>1)*Ncols + c)*2 + (u&1)]
// so a B fragment {W[u0][c], W[u0+1][c]} is ONE contiguous ds_load_b64
// directly into the VGPR pair WMMA consumes (no register shuffles).
//
// WMMA f32 16x16x4 fragment layouts (wave32):
//   A (16x4):  lane m=lane&15 holds row m; g=lane>>4: vgpr0=K(2g), vgpr1=K(2g+1)
//   B (4x16):  lane n=lane&15 holds col n; vgpr0=row(2g), vgpr1=row(2g+1)
//   C/D(16x16): vgpr p, lane&15=col, row = p + 8*(lane>>4)
//
// IR is a template parameter so A-fragment loads get immediate offsets
// (and fuse into one global_load_b64 when IR==1).
// ---------------------------------------------------------------------------
template <int NT, int IR>
__global__ __launch_bounds__(256) void lin_kernel(
    const float* __restrict__ X, const float* __restrict__ Wm,
    float* __restrict__ Y, int K,
    int strideX, int offX, int strideY, int offY, float scale)
{
    extern __shared__ float lw[];            // K * (NT*16) floats
    const int Ncols = NT * 16;
    const int tid = threadIdx.x;

    for (int idx = tid; idx < K * Ncols; idx += 256) {
        const int u = idx / Ncols;
        const int c = idx - u * Ncols;
        lw[(((u >> 1) * Ncols + c) << 1) + (u & 1)] = Wm[idx];
    }
    __syncthreads();

    const int wave = tid >> 5;
    const int lane = tid & 31;
    const int m    = lane & 15;              // row within tile (A) / col (B,C)
    const int g    = lane >> 4;              // half-wave group

    const long long r0 = ((long long)blockIdx.x * 8 + wave) * 16;

    // per-lane A element pointer: starts at X[n, offX + (2g)*IR + i]
    long long r = r0 + m;
    int an = (int)(r / IR);
    int ai = (int)(r - (long long)an * IR);
    const float* ap = X + (long long)an * strideX + offX + ai + 2 * g * IR;

    // per-lane B fragment pointer: k-row-pair g, column m
    const float* bp = lw + 2 * (g * Ncols + m);

    v8f acc[NT] = {};

    for (int kc = 0; kc < K; kc += 4) {
        v2f a;
        if constexpr (IR == 1) {
            a = *(const v2f*)ap;             // one global_load_b64
        } else {
            a.x = ap[0];                     // immediate offsets 0, IR*4
            a.y = ap[IR];
        }
        ap += 4 * IR;
#pragma unroll
        for (int t = 0; t < NT; ++t) {
            v2f b = *(const v2f*)(bp + t * 32);   // ds_load_b64, imm offset
            acc[t] = __builtin_amdgcn_wmma_f32_16x16x4_f32(
                false, a, false, b, (short)0, acc[t], false, false);
        }
        bp += 4 * Ncols;                     // advance two k-row pairs
    }

    // store D: row = r0 + p + 8*g, col = t*16 + m
    long long rs = r0 + 8 * g;
    int sn = (int)(rs / IR);
    int si = (int)(rs - (long long)sn * IR);
#pragma unroll
    for (int p = 0; p < 8; ++p) {
        float* yrow = Y + (long long)sn * strideY + offY + si;
#pragma unroll
        for (int t = 0; t < NT; ++t) {
            const int c = t * 16 + m;
            yrow[c * IR] = acc[t][p] * scale;
        }
        ++si;
        if (si == IR) { si = 0; ++sn; }
    }
}

// ---------------------------------------------------------------------------
// Zero small workspace arrays.
// ---------------------------------------------------------------------------
__global__ void zero_kernel(float* __restrict__ p, int n)
{
    int i = blockIdx.x * 256 + threadIdx.x;
    if (i < n) p[i] = 0.0f;
}

// ---------------------------------------------------------------------------
// Per-column sums / sums-of-squares over the batch (for batchnorm stats).
// Each block reduces a 128-row chunk with coalesced reads; register partials;
// a handful of float atomics per block.
// ---------------------------------------------------------------------------
__global__ __launch_bounds__(256) void colstats_kernel(
    const float* __restrict__ Y, int Wtot,
    float* __restrict__ csum, float* __restrict__ csq)
{
    const int tid = threadIdx.x;
    const long long base = (long long)blockIdx.x * 128;
    float s[3] = {0.f, 0.f, 0.f}, q[3] = {0.f, 0.f, 0.f};
    for (int rr = 0; rr < 128; ++rr) {
        const float* row = Y + (base + rr) * Wtot;
        int c = tid;
        for (int j = 0; c < Wtot; c += 256, ++j) {
            float v = row[c];
            s[j] += v;
            q[j] += v * v;
        }
    }
    int c = tid;
    for (int j = 0; c < Wtot; c += 256, ++j) {
        atomicAdd(&csum[c], s[j]);
        atomicAdd(&csq[c], q[j]);
    }
}

// ---------------------------------------------------------------------------
// Fold column stats -> per-column affine (scale, bias) for one irrep block.
// l==0 (isScalar): center by mean, var = E[x^2]-E[x]^2, bias includes bn_b.
// l>0: norm = mean over (batch, i) of x^2, no centering, no bias.
// ---------------------------------------------------------------------------
__global__ void finalize_kernel(
    const float* __restrict__ csum, const float* __restrict__ csq,
    const float* __restrict__ bnw, const float* __restrict__ bnb,
    float* __restrict__ scol, float* __restrict__ bcol,
    int colOff, int mul, int ir, float invCnt, int isScalar)
{
    int v = blockIdx.x * blockDim.x + threadIdx.x;
    if (v >= mul) return;
    float s = 0.f, q = 0.f;
    for (int i = 0; i < ir; ++i) {
        s += csum[colOff + v * ir + i];
        q += csq[colOff + v * ir + i];
    }
    float mean = s * invCnt;
    float e2   = q * invCnt;
    float var  = isScalar ? (e2 - mean * mean) : e2;
    float sc   = bnw[v] * rsqrtf(var + 1e-5f);
    float b    = isScalar ? (bnb[v] - mean * sc) : 0.f;
    for (int i = 0; i < ir; ++i) {
        scol[colOff + v * ir + i] = sc;
        bcol[colOff + v * ir + i] = b;
    }
}

// ---------------------------------------------------------------------------
// Apply batchnorm affine + gate nonlinearity:
//   cols [0,ns)        -> TANH_C * tanh(t)
//   cols [ns,ns+ng)    -> gate scalars (consumed, not emitted)
//   gated irrep blocks -> t * SIG_C * sigmoid(gate[v])
// ---------------------------------------------------------------------------
__global__ __launch_bounds__(256) void gate_kernel(
    const float* __restrict__ Yin,
    const float* __restrict__ scol, const float* __restrict__ bcol,
    float* __restrict__ H, long long total,
    int Wy, int Wh, int ns, int ng, int mul1, int ir1, int ir2)
{
    long long idx = (long long)blockIdx.x * 256 + threadIdx.x;
    if (idx >= total) return;
    long long n = idx / Wh;
    int c = (int)(idx - n * Wh);
    const float* yrow = Yin + n * Wy;
    float out;
    if (c < ns) {
        float t = yrow[c] * scol[c] + bcol[c];
        out = TANH_C * tanhf(t);
    } else {
        int cg = c - ns;
        int d1 = mul1 * ir1;
        int gidx = (cg < d1) ? (cg / ir1) : (mul1 + (cg - d1) / ir2);
        int src = ns + ng + cg;
        float gin = yrow[ns + gidx] * scol[ns + gidx] + bcol[ns + gidx];
        float gv = SIG_C / (1.0f + __expf(-gin));
        float f = yrow[src] * scol[src] + bcol[src];
        out = f * gv;
    }
    H[n * Wh + c] = out;
}

// ---------------------------------------------------------------------------
// Host-side layer description tables.
// Input tensor order (setup_inputs dict order, flattened):
//  0:x, then per layer k: lin{k}_w*, then bn{k}_w/b interleaved; finally fin_w0/1.
// ---------------------------------------------------------------------------
struct LinDesc { int widx, K, Ncols, ir, offX, offY; };
struct BnDesc  { int wIdx, bIdx, mul, ir, off; };
struct LayerDesc {
    int Wx, Wy, Wh;
    int nLin; LinDesc lin[4];
    int nBn;  BnDesc  bn[4];
    int ns, ng, mul1, ir1, ir2;
};

static const LayerDesc g_layers[4] = {
    // layer 0: in 240 (x), y 576, h 480
    {240, 576, 480, 4,
     {{1, 64, 128, 1, 0, 0}, {2, 64, 96, 1, 0, 128},
      {3, 32, 64, 3, 64, 224}, {4, 16, 32, 5, 160, 416}},
     4,
     {{5, 6, 128, 1, 0}, {7, 8, 96, 1, 128}, {9, -1, 64, 3, 224}, {10, -1, 32, 5, 416}},
     128, 96, 64, 3, 5},
    // layer 1: in 480, y 576, h 480
    {480, 576, 480, 4,
     {{11, 128, 128, 1, 0, 0}, {12, 128, 96, 1, 0, 128},
      {13, 64, 64, 3, 128, 224}, {14, 32, 32, 5, 320, 416}},
     4,
     {{15, 16, 128, 1, 0}, {17, 18, 96, 1, 128}, {19, -1, 64, 3, 224}, {20, -1, 32, 5, 416}},
     128, 96, 64, 3, 5},
    // layer 2
    {480, 576, 480, 4,
     {{21, 128, 128, 1, 0, 0}, {22, 128, 96, 1, 0, 128},
      {23, 64, 64, 3, 128, 224}, {24, 32, 32, 5, 320, 416}},
     4,
     {{25, 26, 128, 1, 0}, {27, 28, 96, 1, 128}, {29, -1, 64, 3, 224}, {30, -1, 32, 5, 416}},
     128, 96, 64, 3, 5},
    // layer 3: in 480, y 192, h 160
    {480, 192, 160, 3,
     {{31, 128, 64, 1, 0, 0}, {32, 128, 32, 1, 0, 64}, {33, 64, 32, 3, 128, 96},
      {0, 0, 0, 0, 0, 0}},
     3,
     {{34, 35, 64, 1, 0}, {36, 37, 32, 1, 64}, {38, -1, 32, 3, 96}, {0, 0, 0, 0, 0}},
     64, 32, 32, 3, 1},
};

static void launch_lin(const float* X, int strideX, float* Y, int strideY,
                       const float* Wm, int K, int Ncols, int ir,
                       int offX, int offY, hipStream_t stream)
{
    const float scale = 1.0f / sqrtf((float)K);
    const long long R = (long long)NBATCH * ir;   // multiple of 128
    dim3 grid((unsigned)(R / 128));
    size_t sh = (size_t)K * Ncols * sizeof(float);

#define LIN_CASE(nt, irr)                                                     \
    lin_kernel<nt, irr><<<grid, 256, sh, stream>>>(                           \
        X, Wm, Y, K, strideX, offX, strideY, offY, scale)

    switch (Ncols / 16) {
    case 8: LIN_CASE(8, 1); break;                       // ir==1 only
    case 6: LIN_CASE(6, 1); break;                       // ir==1 only
    case 4:
        if (ir == 1) LIN_CASE(4, 1);
        else         LIN_CASE(4, 3);
        break;
    default:
        if (ir == 1)      LIN_CASE(2, 1);
        else if (ir == 3) LIN_CASE(2, 3);
        else              LIN_CASE(2, 5);
        break;
    }
#undef LIN_CASE
}

extern "C" void kernel_launch(void* const* d_in, const int* in_sizes, int n_in,
                              void* d_out, int out_size, void* d_ws, size_t ws_size,
                              hipStream_t stream)
{
    (void)in_sizes; (void)n_in; (void)out_size; (void)ws_size;
    const int N = NBATCH;

    // workspace layout
    float* bufY = (float*)d_ws;                       // N * 576
    float* bufH = bufY + (size_t)N * 576;             // N * 480
    float* csum = bufH + (size_t)N * 480;             // 576
    float* csq  = csum + 576;                         // 576
    float* scol = csq  + 576;                         // 576
    float* bcol = scol + 576;                         // 576

    const float* h = (const float*)d_in[0];           // layer-0 input (x)

    for (int k = 0; k < 4; ++k) {
        const LayerDesc& L = g_layers[k];

        // 1) linears (WMMA GEMMs) into bufY
        for (int j = 0; j < L.nLin; ++j) {
            const LinDesc& ld = L.lin[j];
            launch_lin(h, L.Wx, bufY, L.Wy, (const float*)d_in[ld.widx],
                       ld.K, ld.Ncols, ld.ir, ld.offX, ld.offY, stream);
        }

        // 2) batch statistics: zero then column sums / sumsq
        zero_kernel<<<(1152 + 255) / 256, 256, 0, stream>>>(csum, 1152);
        colstats_kernel<<<N / 128, 256, 0, stream>>>(bufY, L.Wy, csum, csq);

        // 3) finalize per irrep block -> per-column scale/bias
        for (int i = 0; i < L.nBn; ++i) {
            const BnDesc& bd = L.bn[i];
            const float* bnw = (const float*)d_in[bd.wIdx];
            const float* bnb = (bd.bIdx >= 0) ? (const float*)d_in[bd.bIdx] : nullptr;
            const float invCnt = 1.0f / ((float)N * (float)bd.ir);
            finalize_kernel<<<1, 128, 0, stream>>>(csum, csq, bnw, bnb, scol, bcol,
                                                   bd.off, bd.mul, bd.ir, invCnt,
                                                   bd.bIdx >= 0 ? 1 : 0);
        }

        // 4) gate -> bufH
        long long total = (long long)N * L.Wh;
        unsigned gblocks = (unsigned)((total + 255) / 256);
        gate_kernel<<<gblocks, 256, 0, stream>>>(bufY, scol, bcol, bufH, total,
                                                 L.Wy, L.Wh, L.ns, L.ng,
                                                 L.mul1, L.ir1, L.ir2);
        h = bufH;
    }

    // final linear: h (160 wide) -> d_out (160 wide)
    float* out = (float*)d_out;
    launch_lin(h, 160, out, 160, (const float*)d_in[39], 64, 64, 1, 0, 0, stream);
    launch_lin(h, 160, out, 160, (const float*)d_in[40], 32, 32, 3, 64, 64, stream);
}